// LearnableWaveletTransform_22187801051640
// MI455X (gfx1250) — compile-verified
//
#include <hip/hip_runtime.h>

typedef __attribute__((ext_vector_type(2))) float v2f;
typedef __attribute__((ext_vector_type(8))) float v8f;

#define TILES_PER_WAVE 8
#define WAVES_PER_BLOCK 4

// db4 decomposition filters, pre-reversed (kGR[t] = DEC_LO[7-t], kHR[t] = DEC_HI[7-t])
// so that  cA[n] = sum_t x_ext[2n+t] * kGR[t],  cD[n] = sum_t x_ext[2n+t] * kHR[t]
// with x_ext[i] = x[sym(i-6)] (symmetric pad by 7, drop first sample).
static constexpr float kGR[8] = {
     0.23037781330885523f,  0.7148465705525415f,   0.6308807679295904f,  -0.02798376941698385f,
    -0.18703481171888114f,  0.030841381835986965f, 0.032883011666982945f, -0.010597401784997278f };
static constexpr float kHR[8] = {
    -0.010597401784997278f, -0.032883011666982945f, 0.030841381835986965f, 0.18703481171888114f,
    -0.02798376941698385f,  -0.6308807679295904f,   0.7148465705525415f,  -0.23037781330885523f };

// ---------------------------------------------------------------------------
// B operand (banded Toeplitz filter matrix) precomputed at C++ compile time.
// Layout matches the V_WMMA f32 4x16 B operand: lane l holds column n = l&15,
// K rows (l>>4)*2 and (l>>4)*2+1 in its two VGPR slots, for 6 chunks of K=4.
// Columns 0..7 = low-pass at relative position n, columns 8..15 = high-pass.
// B[k_abs][n] = filt[k_abs - 2*(n&7)] when the tap index is in [0,8), else 0.
// ---------------------------------------------------------------------------
struct alignas(16) BTab { float v[32][12]; };

static constexpr float tap(int hi, int t) {
    return (t < 0 || t > 7) ? 0.0f : (hi ? kHR[t] : kGR[t]);
}
static constexpr BTab makeBTab() {
    BTab tb{};
    for (int lane = 0; lane < 32; ++lane) {
        const int col16 = lane & 15;
        const int krow  = (lane >> 4) << 1;
        const int hi    = col16 >> 3;
        const int n     = col16 & 7;
        for (int c = 0; c < 6; ++c)
            for (int s = 0; s < 2; ++s)
                tb.v[lane][c * 2 + s] = tap(hi, (4 * c + krow + s) - 2 * n);
    }
    return tb;
}
__device__ __constant__ BTab kBTab = makeBTab();

// ---------------------------------------------------------------------------
// Interior DWT step via fp32 WMMA, persistent waves.
// One wave owns a 16-batch slice and a contiguous strip of TILES_PER_WAVE
// position tiles. Per tile it computes a 16x16 D tile:
//   columns 0..7  = cA at n0..n0+7,  columns 8..15 = cD at n0..n0+7
//   D(16x16) = A(16x24 data window) * B(24x16 Toeplitz)
// as 6 chained V_WMMA_F32_16X16X4_F32 (accumulator-chained in registers).
// ---------------------------------------------------------------------------
__global__ __launch_bounds__(WAVES_PER_BLOCK * 32) void dwt_level_wmma(
    const float* __restrict__ in, int N,
    float* __restrict__ outA, float* __restrict__ outD, int N_out,
    const float* __restrict__ w, const float* __restrict__ bvec,
    int jA, int jD, int n_tiles)
{
    const int lane  = threadIdx.x & 31;
    const int wave  = threadIdx.x >> 5;
    const int gwave = blockIdx.x * WAVES_PER_BLOCK + wave;

    const int t0 = gwave * TILES_PER_WAVE;
    if (t0 >= n_tiles) return;                       // wave-uniform: EXEC stays full
    const int tEnd = (t0 + TILES_PER_WAVE < n_tiles) ? (t0 + TILES_PER_WAVE) : n_tiles;

    const int batch0 = blockIdx.y << 4;
    const int col16  = lane & 15;                    // B/D column; also A row (batch-in-tile)
    const int krow   = (lane >> 4) << 1;             // K slot base: 0 (lanes 0-15) / 2 (16-31)

    // Fetch precomputed constant B operand: 6 chunks of K=4 (12 floats / lane).
    v2f bmat[6];
    {
        const float* bt = kBTab.v[lane];
#pragma unroll
        for (int c = 0; c < 6; ++c)
            bmat[c] = *(const v2f*)(bt + 2 * c);
    }

    // Affine parameters (uniform per wave half-column).
    const bool  hiF = (col16 >= 8);
    const float sA  = (jA >= 0) ? w[jA]    : 1.0f;
    const float tA  = (jA >= 0) ? bvec[jA] : 0.0f;
    const float s   = hiF ? w[jD]    : sA;
    const float t   = hiF ? bvec[jD] : tA;
    float*      outp = hiF ? outD : outA;

    // Input pointer for this lane: window base advances 16 floats per tile.
    const int n0_first = 8 + (t0 << 3);
    const float* rowp = in + (size_t)(batch0 + col16) * N + (2 * n0_first - 6) + krow;

    // Output base: row = batch0 + (lane>=16 ? 8 : 0) + r, col = n0 + (col16&7).
    const int bb = batch0 + ((lane >> 4) << 3);
    float* op = outp + (size_t)bb * N_out + n0_first + (col16 & 7);

#pragma unroll 2
    for (int tt = t0; tt < tEnd; ++tt) {
        __builtin_prefetch(rowp + 128, 0, 0);        // speculative lookahead (global_prefetch_b8)

        v8f acc = {0.f, 0.f, 0.f, 0.f, 0.f, 0.f, 0.f, 0.f};
#pragma unroll
        for (int c = 0; c < 6; ++c) {
            v2f a = *(const v2f*)(rowp + 4 * c);
            acc = __builtin_amdgcn_wmma_f32_16x16x4_f32(
                /*neg_a=*/false, a, /*neg_b=*/false, bmat[c],
                /*c_mod=*/(short)0, acc, /*reuse_a=*/false, /*reuse_b=*/false);
        }

#pragma unroll
        for (int r = 0; r < 8; ++r)
            op[(size_t)r * N_out] = acc[r] * s + t;

        rowp += 16;                                   // next tile: n0 += 8  => window += 16
        op   += 8;
    }
}

// Boundary outputs (symmetric reflection): n in [0,8) and [interior_end, N_out).
__global__ __launch_bounds__(128) void dwt_level_edge(
    const float* __restrict__ in, int N,
    float* __restrict__ outA, float* __restrict__ outD, int N_out,
    const float* __restrict__ w, const float* __restrict__ bvec,
    int jA, int jD, int interior_end, int B, int edge_count)
{
    const int idx = blockIdx.x * blockDim.x + threadIdx.x;
    if (idx >= B * edge_count) return;
    const int e = idx % edge_count;
    const int b = idx / edge_count;
    const int n = (e < 8) ? e : (interior_end + (e - 8));

    const float* row = in + (size_t)b * N;
    float a = 0.f, d = 0.f;
#pragma unroll
    for (int t = 0; t < 8; ++t) {
        int i = 2 * n + t - 6;
        i = (i < 0)  ? (-i - 1)        : i;   // half-sample symmetric reflection
        i = (i >= N) ? (2 * N - 1 - i) : i;
        const float v = row[i];
        a += v * kGR[t];
        d += v * kHR[t];
    }
    const float sA = (jA >= 0) ? w[jA]    : 1.0f;
    const float tA = (jA >= 0) ? bvec[jA] : 0.0f;
    outA[(size_t)b * N_out + n] = a * sA + tA;
    outD[(size_t)b * N_out + n] = d * w[jD] + bvec[jD];
}

extern "C" void kernel_launch(void* const* d_in, const int* in_sizes, int n_in,
                              void* d_out, int out_size, void* d_ws, size_t ws_size,
                              hipStream_t stream) {
    (void)n_in; (void)out_size; (void)ws_size;
    const float* x   = (const float*)d_in[0];
    const float* w   = (const float*)d_in[1];   // wavelet_weights, shape (5,1)
    const float* bb  = (const float*)d_in[2];   // wavelet_bias,    shape (5,1)
    float*       out = (float*)d_out;

    const int B = 256;
    const int T = in_sizes[0] / B;

    // Per-level lengths: len[0]=T input, len[l] = level-l coefficient length.
    int len[5];
    len[0] = T;
    for (int l = 1; l <= 4; ++l) len[l] = (len[l - 1] + 7) / 2;

    // Output layout: [cA4][cD4][cD3][cD2][cD1], each [B, len] row-major.
    size_t offD[5];
    const size_t offA4 = 0;
    offD[4] = (size_t)B * len[4];
    offD[3] = offD[4] + (size_t)B * len[4];
    offD[2] = offD[3] + (size_t)B * len[3];
    offD[1] = offD[2] + (size_t)B * len[2];

    // Ping-pong intermediate cA buffers in workspace (~50 MB).
    float* ws0 = (float*)d_ws;
    float* ws1 = ws0 + (size_t)B * len[1];
    float* bufs[2] = { ws0, ws1 };

    const float* src = x;
    for (int l = 1; l <= 4; ++l) {
        const int N     = len[l - 1];
        const int N_out = len[l];
        float* outA = (l == 4) ? (out + offA4) : bufs[(l - 1) & 1];
        const int jA = (l == 4) ? 0 : -1;       // only the final cA gets its affine
        float* outD = out + offD[l];
        const int jD = 5 - l;                    // cD_l sits at tuple index 5-l

        // Interior tiles: n0 = 8 + 8*t, valid while the 24-wide window stays in-bounds.
        const int n_max   = (N - 16) / 2;
        const int n_tiles = (n_max >= 8) ? ((n_max - 8) / 8 + 1) : 0;
        const int interior_end = 8 + 8 * n_tiles;

        if (n_tiles > 0) {
            const int waves  = (n_tiles + TILES_PER_WAVE - 1) / TILES_PER_WAVE;
            const int blocks = (waves + WAVES_PER_BLOCK - 1) / WAVES_PER_BLOCK;
            dim3 grid(blocks, B / 16);
            dwt_level_wmma<<<grid, WAVES_PER_BLOCK * 32, 0, stream>>>(
                src, N, outA, outD, N_out, w, bb, jA, jD, n_tiles);
        }
        const int edge_count = 8 + (N_out - interior_end);
        const int total      = B * edge_count;
        dwt_level_edge<<<(total + 127) / 128, 128, 0, stream>>>(
            src, N, outA, outD, N_out, w, bb, jA, jD, interior_end, B, edge_count);

        src = outA;
    }
}